// tLNv2_1580547968901
// MI455X (gfx1250) — compile-verified
//
#include <hip/hip_runtime.h>
#include <hip/hip_bf16.h>

// LayerNorm over last dim W=256 of (B=8, C=128, H=128, W=256) fp32, then
// per-channel affine:  out = (x - mean)/sqrt(var+eps) * gain[c] + bias[c].
//
// Memory-bound (256 MiB traffic, ~11.5 us @ 23.3 TB/s). One wave32 per row:
// each lane holds 8 floats as two float4s (coalesced b128, non-temporal),
// mean/variance via 5-step wave32 butterflies (ds_swizzle), scalar-cache
// loads for the wave-uniform gain/bias, NT streaming stores back out.

typedef float v4f __attribute__((ext_vector_type(4)));

#define LN_EPS 1e-8f

__global__ __launch_bounds__(256)
void tln_rows_kernel(const float* __restrict__ inp,
                     const float* __restrict__ gain,
                     const float* __restrict__ bias,
                     float* __restrict__ out,
                     int nrows)
{
    constexpr int W      = 256;   // row length (floats)
    constexpr int LOG2_H = 7;     // H = 128
    constexpr int C_MASK = 127;   // C = 128

    const int lane = threadIdx.x & 31;     // wave32 lane
    const int wave = threadIdx.x >> 5;     // wave within block (0..7)
    const int row  = blockIdx.x * (blockDim.x >> 5) + wave;
    if (row >= nrows) return;

    const v4f* __restrict__ rp = (const v4f*)(inp + (size_t)row * W);
    v4f*       __restrict__ op = (v4f*)(out + (size_t)row * W);

    // Two fully-coalesced 16B loads per lane: lanes 0..31 cover bytes
    // [0,512) then [512,1024) of the row. Non-temporal: single-use stream.
    v4f a = __builtin_nontemporal_load(rp + lane);
    v4f b = __builtin_nontemporal_load(rp + lane + 32);

    // ---- pass 1: mean ----
    float s = ((a.x + a.y) + (a.z + a.w)) + ((b.x + b.y) + (b.z + b.w));
#pragma unroll
    for (int m = 16; m >= 1; m >>= 1)
        s += __shfl_xor(s, m, 32);
    const float mean = s * (1.0f / (float)W);

    // ---- pass 2: variance of centered values (matches reference order) ----
    v4f ca = a - mean;
    v4f cb = b - mean;
    float q = ((ca.x * ca.x + ca.y * ca.y) + (ca.z * ca.z + ca.w * ca.w)) +
              ((cb.x * cb.x + cb.y * cb.y) + (cb.z * cb.z + cb.w * cb.w));
#pragma unroll
    for (int m = 16; m >= 1; m >>= 1)
        q += __shfl_xor(q, m, 32);
    const float var = q * (1.0f / (float)W);
    const float inv = 1.0f / sqrtf(var + LN_EPS);   // full-precision, once/row

    // ---- per-channel affine ----
    // row = (bIdx*C + c)*H + h, H=C=128  ->  c = (row >> 7) & 127.
    // c is wave-uniform: force it scalar so gain/bias come from the
    // scalar constant cache (s_load) instead of 32-lane VMEM.
    const int   c    = __builtin_amdgcn_readfirstlane((row >> LOG2_H) & C_MASK);
    const float g    = gain[c] * inv;
    const float beta = bias[c];

    v4f oa = ca * g + beta;
    v4f ob = cb * g + beta;
    __builtin_nontemporal_store(oa, op + lane);
    __builtin_nontemporal_store(ob, op + lane + 32);
}

extern "C" void kernel_launch(void* const* d_in, const int* in_sizes, int n_in,
                              void* d_out, int out_size, void* d_ws, size_t ws_size,
                              hipStream_t stream)
{
    const float* inp  = (const float*)d_in[0];   // (8,128,128,256) fp32
    const float* gain = (const float*)d_in[1];   // (1,128,1,1)
    const float* bias = (const float*)d_in[2];   // (1,128,1,1)
    float*       out  = (float*)d_out;

    const int W = 256;
    const int nrows = in_sizes[0] / W;           // 8*128*128 = 131072

    const int wavesPerBlock = 8;                 // 256 threads = 8 wave32
    dim3 block(256);
    dim3 grid((nrows + wavesPerBlock - 1) / wavesPerBlock);
    tln_rows_kernel<<<grid, block, 0, stream>>>(inp, gain, bias, out, nrows);
}